// DemoDynamicTokenSorterV2_20478404068046
// MI455X (gfx1250) — compile-verified
//
#include <hip/hip_runtime.h>

// Problem constants (match reference)
#define NTOK 6400
#define BB   8
#define DD   2048
#define TGc  4
#define HGc  40
#define WGc  40
#define P_LOW  0.2f
#define P_HIGH 0.6f
#define EPSc   1e-6f

// Workspace dword layout (needs ~130KB of d_ws)
#define H_NK 0
#define H_M  1
#define H_C  2
#define OFF_KEEP (8)
#define OFF_MIDX (8 + NTOK)
#define OFF_MROW (8 + 2*NTOK)
#define OFF_MSCL (8 + 3*NTOK)
#define OFF_MCL  (8 + 4*NTOK)
#define OFF_COFF (8 + 5*NTOK)

typedef float v8f __attribute__((ext_vector_type(8)));
typedef float v2f __attribute__((ext_vector_type(2)));

// ---------------------------------------------------------------------------
// Kernel 1: build the static index plan (single block, deterministic).
// keep_idx (ascending), CSR of merge tokens grouped by compacted cluster id,
// per-entry scale w/denom, cluster offsets. Mirrors reference plan() exactly.
// ---------------------------------------------------------------------------
__global__ __launch_bounds__(256) void plan_kernel(const float* __restrict__ tmpl,
                                                   int* wsI, float* wsF) {
  __shared__ int   sMapped[NTOK];      // cluster id -> mapped id per merge token
  __shared__ float sW[NTOK];           // raw merge weight per merge token
  __shared__ int   scanK[257], scanM[257];
  __shared__ int   pres[321], cmap[321], cnt[321], coff[322];
  __shared__ float sden[321];
  __shared__ int   sh_nk, sh_m, sh_C, sh_nt, sh_nh, sh_nw, sh_P;

  const int tid = threadIdx.x;
  const int chunk = NTOK / 256;              // 25, exact
  const int beg = tid * chunk, end = beg + chunk;

  // --- stable chunked compaction of keep / merge indices ---
  int ck = 0, cm = 0;
  for (int i = beg; i < end; ++i) {
    float p = tmpl[i];
    if (p >= P_HIGH) ck++;
    else if (p >= P_LOW) cm++;
  }
  scanK[tid] = ck; scanM[tid] = cm;
  __syncthreads();
  if (tid == 0) {
    int a = 0, b = 0;
    for (int t = 0; t < 256; ++t) {
      int k = scanK[t], m = scanM[t];
      scanK[t] = a; scanM[t] = b; a += k; b += m;
    }
    sh_nk = a; sh_m = b;
  }
  __syncthreads();
  int ko = scanK[tid], mo = scanM[tid];
  for (int i = beg; i < end; ++i) {
    float p = tmpl[i];
    if (p >= P_HIGH) {
      wsI[OFF_KEEP + ko++] = i;
    } else if (p >= P_LOW) {
      wsI[OFF_MIDX + mo] = i;
      float w = (p - P_LOW) / (P_HIGH - P_LOW);
      sW[mo] = fminf(fmaxf(w, 0.f), 1.f);
      mo++;
    }
  }
  __syncthreads();

  const int m = sh_m;
  // --- grid-bin search (tiny, thread 0) ---
  if (tid == 0) {
    if (m < 8) { sh_C = 0; sh_P = 0; sh_nt = 1; sh_nh = 1; sh_nw = 1; }
    else {
      int target = 320;                       // ceil(N*0.05)
      int t2 = (m + 9) / 10; if (t2 < 1) t2 = 1;
      if (t2 < target) target = t2;
      if (m < target) target = m;
      int bp = 1, bt = 1, bh = 1, bw = 1;
      for (int nt = 1; nt <= TGc; ++nt)
        for (int nh = 1; nh <= HGc; ++nh) {
          int mx = target / (nt * nh); if (mx > WGc) mx = WGc;
          if (mx < 1) continue;
          int prod = nt * nh * mx;
          if (prod > bp) { bp = prod; bt = nt; bh = nh; bw = mx; }
        }
      sh_nt = bt; sh_nh = bh; sh_nw = bw; sh_P = bt * bh * bw;  // P <= target <= 320
    }
  }
  __syncthreads();

  if (m >= 8) {
    const int nt = sh_nt, nh = sh_nh, nw = sh_nw, P = sh_P;
    for (int c = tid; c < P; c += 256) pres[c] = 0;
    __syncthreads();
    for (int j = tid; j < m; j += 256) {
      int g = wsI[OFF_MIDX + j];
      int t = g / (HGc * WGc), h = (g / WGc) % HGc, w = g % WGc;
      int bt = t * nt / TGc; if (bt > nt - 1) bt = nt - 1;
      int bh = h * nh / HGc; if (bh > nh - 1) bh = nh - 1;
      int bw = w * nw / WGc; if (bw > nw - 1) bw = nw - 1;
      int cid = bt * (nh * nw) + bh * nw + bw;
      sMapped[j] = cid;
      pres[cid] = 1;
    }
    __syncthreads();
    if (tid == 0) {                         // unique(sorted) rank == prefix of presence
      int r = 0;
      for (int c = 0; c < P; ++c) cmap[c] = pres[c] ? r++ : -1;
      sh_C = r;
    }
    __syncthreads();
    for (int j = tid; j < m; j += 256) sMapped[j] = cmap[sMapped[j]];
    __syncthreads();
    const int C = sh_C;
    for (int c = tid; c < C; c += 256) {    // deterministic per-cluster denom + count
      float ds = 0.f; int n = 0;
      for (int j = 0; j < m; ++j) if (sMapped[j] == c) { ds += sW[j]; n++; }
      sden[c] = fmaxf(ds, EPSc); cnt[c] = n;
    }
    __syncthreads();
    if (tid == 0) {
      int a = 0;
      for (int c = 0; c < C; ++c) { coff[c] = a; a += cnt[c]; }
      coff[C] = a;
    }
    __syncthreads();
    for (int c = tid; c < C; c += 256) {    // fill CSR (ascending j -> stable)
      int pos = coff[c]; float d = sden[c];
      for (int j = 0; j < m; ++j) if (sMapped[j] == c) {
        wsI[OFF_MROW + pos] = wsI[OFF_MIDX + j];
        wsF[OFF_MSCL + pos] = sW[j] / d;
        wsI[OFF_MCL  + pos] = c;
        pos++;
      }
    }
    __syncthreads();
    for (int c = tid; c <= C; c += 256) wsI[OFF_COFF + c] = coff[c];
  } else {
    if (tid == 0) wsI[OFF_COFF + 0] = 0;
  }
  __syncthreads();
  if (tid == 0) { wsI[H_NK] = sh_nk; wsI[H_M] = sh_m; wsI[H_C] = sh_C; }
}

// ---------------------------------------------------------------------------
// Kernel 2: keep-row gather as a VGPR-free async copy:
// global -> LDS (ASYNCcnt) -> global, 8KB row per 256-thread block.
// `stage` is the only LDS object, so its LDS offset is 0.
// ---------------------------------------------------------------------------
__global__ __launch_bounds__(256) void gather_keep(const float* __restrict__ hid,
                                                   const int* __restrict__ wsI,
                                                   float* __restrict__ out, int K) {
  __shared__ char stage[8192];
  const int k = blockIdx.x % K;
  const int b = blockIdx.x / K;
  const int nk = wsI[H_NK];
  if (k >= nk) return;                      // uniform per block
  const int srow = wsI[OFF_KEEP + k];
  const char* s = (const char*)(hid + ((size_t)b * NTOK + srow) * DD);
  char*       d = (char*)(out + ((size_t)b * K + k) * DD);
  unsigned l0 = threadIdx.x * 16u;
  unsigned l1 = l0 + 4096u;
  asm volatile("" :: "v"(&stage[0]) : "memory");   // keep LDS allocation alive
  asm volatile("global_load_async_to_lds_b128 %0, %1, off" :: "v"(l0), "v"(s + l0) : "memory");
  asm volatile("global_load_async_to_lds_b128 %0, %1, off" :: "v"(l1), "v"(s + l1) : "memory");
  asm volatile("s_wait_asynccnt 0x0" ::: "memory");
  asm volatile("global_store_async_from_lds_b128 %0, %1, off" :: "v"(d + l0), "v"(l0) : "memory");
  asm volatile("global_store_async_from_lds_b128 %0, %1, off" :: "v"(d + l1), "v"(l1) : "memory");
  asm volatile("s_wait_asynccnt 0x0" ::: "memory");
}

// ---------------------------------------------------------------------------
// Kernel 3: cluster rows via V_WMMA_F32_16X16X4_F32.
// One wave per (16-cluster tile, 128-col group, batch). A = 16x4 scale tile
// (nonzero only at the token's cluster row), B = 4 token rows x 16 cols.
// Layout rule (ISA 7.12.2): element index = vgpr + nVGPR*(lane>=16).
// Tail handling is branchless: OOB chunk slots are clamped to a valid CSR
// entry and their A-weight forced to 0, so EXEC stays all-ones (WMMA req.)
// and the hot loop has no exec-divergent branches.
// ---------------------------------------------------------------------------
__global__ __launch_bounds__(32) void merge_wmma(const float* __restrict__ hid,
                                                 const int* __restrict__ wsI,
                                                 const float* __restrict__ wsF,
                                                 float* __restrict__ out, int K) {
  const int C = wsI[H_C];
  const int c0 = blockIdx.x * 16;
  if (c0 >= C) return;
  const int nk = wsI[H_NK];
  const int b = blockIdx.z;
  const int colbase = blockIdx.y * 128;
  const int lane = threadIdx.x;
  const int half = lane >> 4;               // K pair select
  const int nlo  = lane & 15;               // M (A) / N (B,D) index
  const int jbeg = wsI[OFF_COFF + c0];
  int cend = c0 + 16; if (cend > C) cend = C;
  const int jend = wsI[OFF_COFF + cend];

  v8f acc[8];
  const v8f vzero = {0.f,0.f,0.f,0.f,0.f,0.f,0.f,0.f};
#pragma unroll
  for (int t = 0; t < 8; ++t) acc[t] = vzero;

  const size_t bbase = (size_t)b * NTOK * DD;
  for (int j = jbeg; j < jend; j += 4) {
    if (j + 4 < jend)                        // gfx1250 global_prefetch
      __builtin_prefetch(hid + bbase + (size_t)wsI[OFF_MROW + j + 4] * DD + colbase, 0, 0);
    const int ja = j + half * 2;             // this lane supplies K = 2*half, 2*half+1
    const int jb = ja + 1;
    const bool vA = ja < jend;
    const bool vB = jb < jend;
    const int jac = vA ? ja : jbeg;          // clamp -> unconditional loads
    const int jbc = vB ? jb : jbeg;
    const int   rowA = wsI[OFF_MROW + jac];
    const float sA   = wsF[OFF_MSCL + jac];
    const int   clA  = wsI[OFF_MCL  + jac];
    const int   rowB = wsI[OFF_MROW + jbc];
    const float sB   = wsF[OFF_MSCL + jbc];
    const int   clB  = wsI[OFF_MCL  + jbc];

    v2f a;
    a.x = (vA && clA == c0 + nlo) ? sA : 0.f;  // A[M=nlo][K=2*half]
    a.y = (vB && clB == c0 + nlo) ? sB : 0.f;  // A[M=nlo][K=2*half+1]

    const float* hA = hid + bbase + (size_t)rowA * DD + colbase + nlo;
    const float* hB = hid + bbase + (size_t)rowB * DD + colbase + nlo;
#pragma unroll
    for (int t = 0; t < 8; ++t) {
      v2f bv;
      bv.x = hA[t * 16];                     // B[K=2*half][N=nlo]
      bv.y = hB[t * 16];                     // B[K=2*half+1][N=nlo]
      acc[t] = __builtin_amdgcn_wmma_f32_16x16x4_f32(
          false, a, false, bv, (short)0, acc[t], false, false);
    }
  }

  // D layout: element i of acc -> row M = i + 8*half, col N = nlo
  float* orow = out + ((size_t)b * K + nk) * DD + colbase + nlo;
#pragma unroll
  for (int t = 0; t < 8; ++t) {
#pragma unroll
    for (int i = 0; i < 8; ++i) {
      int row = c0 + i + half * 8;
      if (row < C) orow[(size_t)row * DD + t * 16] = acc[t][i];
    }
  }
}

extern "C" void kernel_launch(void* const* d_in, const int* in_sizes, int n_in,
                              void* d_out, int out_size, void* d_ws, size_t ws_size,
                              hipStream_t stream) {
  (void)in_sizes; (void)n_in; (void)ws_size;
  const float* hidden = (const float*)d_in[0];     // (B, N, D) f32
  const float* tmpl   = (const float*)d_in[2];     // (N,) f32
  float* out = (float*)d_out;
  int*   wsI = (int*)d_ws;
  float* wsF = (float*)d_ws;
  const int K = out_size / (BB * DD);              // exact K = nk + C

  plan_kernel<<<1, 256, 0, stream>>>(tmpl, wsI, wsF);
  gather_keep<<<BB * K, 256, 0, stream>>>(hidden, wsI, out, K);
  dim3 g3((unsigned)((K + 15) / 16), 16u, (unsigned)BB);
  merge_wmma<<<g3, 32, 0, stream>>>(hidden, wsI, wsF, out, K);
}